// GCNLayer_1194000908631
// MI455X (gfx1250) — compile-verified
//
#include <hip/hip_runtime.h>

#define N_NODES 10000
#define N_EDGES 640000
#define D 128

typedef __attribute__((ext_vector_type(2))) float v2f;
typedef __attribute__((ext_vector_type(8))) float v8f;

// ---------------------------------------------------------------------------
// Phase 0: zero the accumulator h[N_NODES][D] in workspace
// ---------------------------------------------------------------------------
__global__ __launch_bounds__(256) void gcn_zero(float* __restrict__ h, int n) {
  int i = blockIdx.x * blockDim.x + threadIdx.x;
  if (i < n) h[i] = 0.0f;
}

// ---------------------------------------------------------------------------
// Phase 1: gather feature[src[e]] and atomically sum into h[dst[e]].
// One wave per edge: 32 lanes x float4 = 128 floats = one full row.
// Reads are fully coalesced (512B/wave); feature (5.12MB) and h (5.12MB)
// stay L2-resident, atomics resolve at L2.
// ---------------------------------------------------------------------------
__global__ __launch_bounds__(256) void gcn_scatter(
    const float* __restrict__ feature,
    const long long* __restrict__ src,
    const long long* __restrict__ dst,
    float* __restrict__ h) {
  int t = blockIdx.x * blockDim.x + threadIdx.x;
  int e = t >> 5;          // edge index (wave-uniform)
  int lane = t & 31;       // lane -> 4-float chunk of the row
  if (e >= N_EDGES) return;

  long long s = src[e];
  long long d = dst[e];

  const float4 v =
      reinterpret_cast<const float4*>(feature + s * (long long)D)[lane];
  float* hd = h + d * (long long)D + lane * 4;
  atomicAdd(hd + 0, v.x);   // -> global_atomic_add_f32 (no return)
  atomicAdd(hd + 1, v.y);
  atomicAdd(hd + 2, v.z);
  atomicAdd(hd + 3, v.w);
}

// ---------------------------------------------------------------------------
// Phase 2: out = h @ W + b using V_WMMA_F32_16X16X4_F32.
// One wave per 16x16 output tile; 625 row-tiles x 8 col-tiles = 5000 tiles.
// 8 waves / block -> 625 blocks of 256 threads. K=128 fully unrolled
// (32 WMMA ops per tile).
//
// VGPR layouts (ISA 7.12.2, 32-bit):
//   A 16x4 : lane<16 -> {A[m][k+0], A[m][k+1]}, lane>=16 -> {A[m][k+2], A[m][k+3]}
//   B 4x16 : lane<16 -> {B[k+0][n], B[k+1][n]}, lane>=16 -> {B[k+2][n], B[k+3][n]}
//   C 16x16: 8 VGPRs, lane<16 -> (M=v, N=lane), lane>=16 -> (M=v+8, N=lane-16)
// ---------------------------------------------------------------------------
__global__ __launch_bounds__(256) void gcn_gemm_wmma(
    const float* __restrict__ h,
    const float* __restrict__ W,
    const float* __restrict__ bias,
    float* __restrict__ out) {
  const int lane = threadIdx.x & 31;
  const int wave = threadIdx.x >> 5;
  const int tile = blockIdx.x * 8 + wave;
  const int tileRow = tile >> 3;   // 0..624
  const int tileCol = tile & 7;    // 0..7
  if (tileRow >= N_NODES / 16) return;

  const int m  = lane & 15;          // A row in tile / B column in tile
  const int kh = (lane >> 4) << 1;   // 0 for lanes 0-15, 2 for lanes 16-31

  // A: h[(tileRow*16+m)][k+kh .. k+kh+1]  (8B aligned: kh,k even)
  const float* arow = h + (tileRow * 16 + m) * D + kh;
  // B: W[(k+kh .. k+kh+1)][tileCol*16+m]
  const float* bcol = W + tileCol * 16 + m;

  v8f c = {};
#pragma unroll
  for (int k = 0; k < D; k += 4) {
    const float2 af = *reinterpret_cast<const float2*>(arow + k);
    v2f a; a.x = af.x; a.y = af.y;
    v2f b;
    b.x = bcol[(k + kh + 0) * D];
    b.y = bcol[(k + kh + 1) * D];
    // (neg_a, A, neg_b, B, c_mod, C, reuse_a, reuse_b)
    c = __builtin_amdgcn_wmma_f32_16x16x4_f32(false, a, false, b,
                                              (short)0, c, false, false);
  }

  // Write D + bias. lane<16: N=lane, M=v; lane>=16: N=lane-16, M=v+8.
  const int n = lane & 15;
  const int mbase = (lane >> 4) << 3;
  const float bv = bias[tileCol * 16 + n];
  float* ocol = out + tileCol * 16 + n;
#pragma unroll
  for (int v = 0; v < 8; ++v) {
    ocol[(tileRow * 16 + mbase + v) * D] = c[v] + bv;
  }
}

// ---------------------------------------------------------------------------
extern "C" void kernel_launch(void* const* d_in, const int* in_sizes, int n_in,
                              void* d_out, int out_size, void* d_ws,
                              size_t ws_size, hipStream_t stream) {
  const float* feature   = (const float*)d_in[0];
  const long long* src   = (const long long*)d_in[1];  // jnp.int64
  const long long* dst   = (const long long*)d_in[2];  // jnp.int64
  const float* W         = (const float*)d_in[3];
  const float* b         = (const float*)d_in[4];
  float* out             = (float*)d_out;
  float* h               = (float*)d_ws;               // 10000*128*4 = 5.12 MB

  const int hElems = N_NODES * D;
  gcn_zero<<<(hElems + 255) / 256, 256, 0, stream>>>(h, hElems);

  const long long scatterThreads = (long long)N_EDGES * 32;
  gcn_scatter<<<(unsigned)((scatterThreads + 255) / 256), 256, 0, stream>>>(
      feature, src, dst, h);

  gcn_gemm_wmma<<<(N_NODES / 16) * 8 / 8, 256, 0, stream>>>(h, W, b, out);
}